// BoxMatcher_87857851007546
// MI455X (gfx1250) — compile-verified
//
#include <hip/hip_runtime.h>
#include <hip/hip_bf16.h>
#include <cstdint>

#define NBATCH 32
#define NPRED  2048
#define NTGT   128
#define BIGF   1e30f

typedef float v2f  __attribute__((ext_vector_type(2)));
typedef float v8f  __attribute__((ext_vector_type(8)));
typedef unsigned int u32x4 __attribute__((ext_vector_type(4)));
typedef int   i32x4 __attribute__((ext_vector_type(4)));
typedef int   i32x8 __attribute__((ext_vector_type(8)));

// ---------------------------------------------------------------------------
// Phase 1: cost matrix, stored transposed as costT[b][tgt][pred] (fp32).
// One block = (batch b, 16-wide pred tile). 8 waves; wave w owns tgt tile w.
// Each wave computes a 16(tgt) x 16(pred) tile:
//   - rank-2 union base (area_t + area_p) via V_WMMA_F32_16X16X4_F32
//   - L1 / IoU / GIoU terms elementwise on the VALU (co-executes with WMMA)
// Target boxes (128 x float4 = 2KB) staged to LDS via the Tensor Data Mover.
// ---------------------------------------------------------------------------
__global__ __launch_bounds__(256)
void cost_kernel(const float* __restrict__ outB,   // [B, NPRED, 4]
                 const float* __restrict__ tgtB,   // [B, NTGT, 4]
                 float* __restrict__ costT)        // [B, NTGT, NPRED]
{
    __shared__ float4 sTgt[NTGT];

    const int b     = blockIdx.x;
    const int tileP = blockIdx.y;           // 0..127
    const int wave  = threadIdx.x >> 5;     // 0..7  -> tgt tile
    const int lane  = threadIdx.x & 31;

#if __has_builtin(__builtin_amdgcn_tensor_load_to_lds)
    if (threadIdx.x == 0) {
        // D# describing a 1-D tensor of 512 fp32 (the 128 tgt boxes of batch b)
        uint32_t lds_addr = (uint32_t)(uintptr_t)(&sTgt[0]);   // low 32 bits = LDS offset
        uint64_t gaddr    = (uint64_t)(uintptr_t)(tgtB + (size_t)b * NTGT * 4);
        u32x4 g0; i32x8 g1; i32x4 g2; i32x4 g3;
        g0[0] = 1u;                                            // count=1, user descriptor
        g0[1] = lds_addr;                                      // lds_addr [63:32]
        g0[2] = (uint32_t)(gaddr & 0xFFFFFFFFull);             // global_addr lo
        g0[3] = (uint32_t)((gaddr >> 32) & 0x01FFFFFFull) | (2u << 30); // addr hi | type=2
        g1[0] = (int)(2u << 16);          // data_size=2 (4B), no multicast/pad/iterate
        g1[1] = (int)(512u << 16);        // tensor_dim0 = 512 (lo16 in [31:16])
        g1[2] = (int)(1u   << 16);        // tensor_dim0 hi = 0 | tensor_dim1 = 1
        g1[3] = (int)(512u << 16);        // tensor_dim1 hi = 0 | tile_dim0 = 512
        g1[4] = 1;                        // tile_dim1 = 1, tile_dim2 = 0
        g1[5] = 512;                      // tensor_dim0_stride lo
        g1[6] = (int)(512u << 16);        // stride0 hi = 0 | tensor_dim1_stride lo = 512
        g1[7] = 0;
        g2 = (i32x4){0, 0, 0, 0};
        g3 = (i32x4){0, 0, 0, 0};
#if __clang_major__ >= 23
        __builtin_amdgcn_tensor_load_to_lds(g0, g1, g2, g3, (i32x8){0,0,0,0,0,0,0,0}, 0);
#else
        __builtin_amdgcn_tensor_load_to_lds(g0, g1, g2, g3, 0);
#endif
    }
    __builtin_amdgcn_s_wait_tensorcnt(0);
#else
#warning "CDNA5 TDM builtin unavailable in device pass: using cooperative-load fallback"
    for (int t = threadIdx.x; t < NTGT; t += blockDim.x)
        sTgt[t] = ((const float4*)tgtB)[(size_t)b * NTGT + t];
#endif
    __syncthreads();

    const int p  = tileP * 16 + (lane & 15);
    const int hi = lane >> 4;
    const float4 pb = ((const float4*)outB)[(size_t)b * NPRED + p];

    // WMMA operands: A[t,0]=area_t, A[t,1]=1 (K=2,3 zero);  B[0,p]=1, B[1,p]=area_p
    float areaT = 0.f, areaP = 0.f, oneA = 0.f, oneB = 0.f;
    if (hi == 0) {
        float4 tb0 = sTgt[wave * 16 + lane];
        areaT = (tb0.z - tb0.x) * (tb0.w - tb0.y);
        areaP = (pb.z - pb.x) * (pb.w - pb.y);
        oneA = 1.f; oneB = 1.f;
    }
    v2f Am; Am.x = areaT; Am.y = oneA;
    v2f Bm; Bm.x = oneB;  Bm.y = areaP;
    v8f C0 = {};
    v8f D = __builtin_amdgcn_wmma_f32_16x16x4_f32(
        /*neg_a=*/false, Am, /*neg_b=*/false, Bm,
        /*c_mod=*/(short)0, C0, /*reuse_a=*/false, /*reuse_b=*/false);
    // D[r] = area_t[wave*16 + r + 8*hi] + area_p[p]

    float* baseRow = costT + ((size_t)b * NTGT + wave * 16 + hi * 8) * NPRED + p;
    #pragma unroll
    for (int r = 0; r < 8; ++r) {
        float4 tb = sTgt[wave * 16 + hi * 8 + r];
        float sumA = D[r];
        float l1 = fabsf(pb.x - tb.x) + fabsf(pb.y - tb.y)
                 + fabsf(pb.z - tb.z) + fabsf(pb.w - tb.w);
        float iw = fmaxf(fminf(pb.z, tb.z) - fmaxf(pb.x, tb.x), 0.f);
        float ih = fmaxf(fminf(pb.w, tb.w) - fmaxf(pb.y, tb.y), 0.f);
        float inter = iw * ih;
        float uni   = sumA - inter;
        float iou   = inter / (uni + 1e-8f);
        float ew = fmaxf(fmaxf(pb.z, tb.z) - fminf(pb.x, tb.x), 0.f);
        float eh = fmaxf(fmaxf(pb.w, tb.w) - fminf(pb.y, tb.y), 0.f);
        float ea = ew * eh;
        float giou = iou - (ea - uni) / (ea + 1e-8f);
        baseRow[(size_t)r * NPRED] = l1 - giou;
    }
}

// ---------------------------------------------------------------------------
// Phase 2: Jonker-Volgenant LSAP, one block (256 thr) per batch.
// Rows = targets (128), cols = preds (2048). Thread t owns columns
// [8t, 8t+8): v / spc / path / scanned-mask live in REGISTERS during a
// phase (single-owner state). Inner step = 2x global b128 row loads +
// wave32 shfl_xor butterfly argmin + 2 barriers. spc/path are dumped to
// LDS once per phase for the cross-thread dual update & augmentation.
// ---------------------------------------------------------------------------
__global__ __launch_bounds__(256)
void lsap_kernel(const float* __restrict__ costT, float* __restrict__ outF)
{
    const int b    = blockIdx.x;
    const int tid  = threadIdx.x;
    const int lane = tid & 31;
    const int wv   = tid >> 5;

    __shared__ float spcL[NPRED];
    __shared__ int   pathL[NPRED];
    __shared__ int   row4col[NPRED];
    __shared__ float u_[NTGT];
    __shared__ int   col4row[NTGT];
    __shared__ unsigned char SR[NTGT];
    __shared__ float rvalW[8];
    __shared__ int   ridxW[8];
    __shared__ int   sh_cur, sh_sink, sh_j;
    __shared__ float sh_minv;

    float vreg[8];                         // dual variable v for owned columns
    #pragma unroll
    for (int k = 0; k < 8; ++k) vreg[k] = 0.f;
    for (int j = tid; j < NPRED; j += 256) row4col[j] = -1;
    for (int t = tid; t < NTGT;  t += 256) { u_[t] = 0.f; col4row[t] = -1; }
    __syncthreads();

    const float* costB = costT + (size_t)b * NTGT * NPRED;

    for (int i = 0; i < NTGT; ++i) {
        if (i + 1 < NTGT)
            __builtin_prefetch(costB + (size_t)(i + 1) * NPRED + tid * 8, 0, 1);

        float spcreg[8];
        int   pathreg[8];
        unsigned scmask = 0u;
        #pragma unroll
        for (int k = 0; k < 8; ++k) { spcreg[k] = BIGF; pathreg[k] = -1; }
        for (int t = tid; t < NTGT; t += 256) SR[t] = 0;
        if (tid == 0) { sh_cur = i; sh_minv = 0.f; sh_sink = -1; }
        __syncthreads();

        int   cur  = i;
        float minv = 0.f;
        while (true) {
            if (tid == 0) SR[cur] = 1;
            const float ucur = u_[cur];
            const float4* crow4 = (const float4*)(costB + (size_t)cur * NPRED);
            const float4 c0 = crow4[tid * 2];
            const float4 c1 = crow4[tid * 2 + 1];
            const float cols[8] = { c0.x, c0.y, c0.z, c0.w, c1.x, c1.y, c1.z, c1.w };

            float bestv = BIGF; int besti = NPRED;
            #pragma unroll
            for (int k = 0; k < 8; ++k) {
                if (!((scmask >> k) & 1u)) {
                    float red = minv + cols[k] - ucur - vreg[k];
                    if (red < spcreg[k]) { spcreg[k] = red; pathreg[k] = cur; }
                    float sp = spcreg[k];
                    if (sp < bestv) { bestv = sp; besti = (tid << 3) + k; } // k asc => first idx
                }
            }
            // wave32 butterfly (val, idx) lexicographic min
            #pragma unroll
            for (int m = 16; m > 0; m >>= 1) {
                float ov = __shfl_xor(bestv, m, 32);
                int   oi = __shfl_xor(besti, m, 32);
                if (ov < bestv || (ov == bestv && oi < besti)) { bestv = ov; besti = oi; }
            }
            if (lane == 0) { rvalW[wv] = bestv; ridxW[wv] = besti; }
            __syncthreads();
            if (tid == 0) {
                float bv = rvalW[0]; int bi = ridxW[0];
                #pragma unroll
                for (int w = 1; w < 8; ++w) {
                    float ov = rvalW[w]; int oi = ridxW[w];
                    if (ov < bv || (ov == bv && oi < bi)) { bv = ov; bi = oi; }
                }
                sh_minv = bv; sh_j = bi;
                int rc = row4col[bi];
                sh_sink = (rc < 0) ? bi : -1;
                sh_cur  = rc;
            }
            __syncthreads();
            const int js = sh_j;
            if ((js >> 3) == tid) scmask |= 1u << (js & 7);   // mark scanned (owner)
            minv = sh_minv;
            if (sh_sink >= 0) break;
            cur = sh_cur;
        }

        // Phase end: publish per-column state, apply dual updates, augment.
        #pragma unroll
        for (int k = 0; k < 8; ++k) {
            int j = (tid << 3) + k;
            spcL[j]  = spcreg[k];
            pathL[j] = pathreg[k];
            if ((scmask >> k) & 1u) vreg[k] += spcreg[k] - minv;   // v update (owner)
        }
        __syncthreads();
        for (int t = tid; t < NTGT; t += 256) {
            if (t == i)     u_[t] += minv;
            else if (SR[t]) u_[t] += minv - spcL[col4row[t]];
        }
        __syncthreads();
        if (tid == 0) {
            int j = sh_sink;
            while (true) {
                int ii = pathL[j];
                row4col[j] = ii;
                int nj = col4row[ii];
                col4row[ii] = j;
                if (ii == i) break;
                j = nj;
            }
        }
        __syncthreads();
    }

    // Emit pairs sorted by pred index: rank = #{t' : col4row[t'] < col4row[t]}
    for (int t = tid; t < NTGT; t += 256) {
        int pv = col4row[t];
        int rank = 0;
        for (int t2 = 0; t2 < NTGT; ++t2) rank += (col4row[t2] < pv) ? 1 : 0;
        outF[(size_t)b * 2 * NTGT + rank]        = (float)pv;   // pred_idx
        outF[(size_t)b * 2 * NTGT + NTGT + rank] = (float)t;    // tgt_idx
    }
}

extern "C" void kernel_launch(void* const* d_in, const int* in_sizes, int n_in,
                              void* d_out, int out_size, void* d_ws, size_t ws_size,
                              hipStream_t stream)
{
    const float* outB = (const float*)d_in[0];   // [32, 2048, 4]
    const float* tgtB = (const float*)d_in[1];   // [32, 128, 4]
    float* costT = (float*)d_ws;                 // 32*128*2048 fp32 = 32 MB scratch

    dim3 gridC(NBATCH, NPRED / 16);              // (batch, pred tile)
    cost_kernel<<<gridC, 256, 0, stream>>>(outB, tgtB, costT);
    lsap_kernel<<<NBATCH, 256, 0, stream>>>(costT, (float*)d_out);
}